// WVP_se_26121991094406
// MI455X (gfx1250) — compile-verified
//
#include <hip/hip_runtime.h>
#include <math.h>

#define HS     64
#define DQ     1039
#define DH     1028
#define DIN    3106
#define EMBW   2067    // DH + DQ: columns of W1 multiplying the broadcast emb
#define HQ     79
#define NB     16
#define NN     1024
#define NHEADS 24
#define MLW    4
#define NEGV   (-1.0e10f)

typedef __attribute__((ext_vector_type(2))) float v2f;
typedef __attribute__((ext_vector_type(8))) float v8f;

// workspace layout (float offsets)
#define OFF_G2   0          // 16 * 1040 (g2[b,d], padded row 1040)
#define OFF_C0   16640      // 16
#define OFF_LOG  16656      // 16 * 1024  attn_logits
#define OFF_ADD  33040      // 16 * 128   per-batch rank-1 update (s|e)
#define OFF_BASE 35200      // 16384 * 128 base activations (s:0-63, e:64-127)

// ---------------------------------------------------------------------------
// Kernel 0: ctx_g = ctx@U1^T + U1_b ; g2 = ctx_g@V1 ; c0 = ctx_g . V1_b
// ---------------------------------------------------------------------------
__global__ __launch_bounds__(256) void init_kernel(
    const float* __restrict__ wenc_n, const float* __restrict__ U1_W,
    const float* __restrict__ U1_b, const float* __restrict__ V1_W,
    const float* __restrict__ V1_b, float* __restrict__ ws)
{
  __shared__ float cg[HQ];
  const int b = blockIdx.x, tid = threadIdx.x;
  const float* ctx = wenc_n + (size_t)b * NN * DQ;   // n = 0 row
  if (tid < HQ) {
    float acc = U1_b[tid];
    const float* u = U1_W + (size_t)tid * DQ;
    for (int d = 0; d < DQ; ++d) acc = fmaf(ctx[d], u[d], acc);
    cg[tid] = acc;
  }
  __syncthreads();
  for (int d = tid; d < 1040; d += blockDim.x) {
    float acc = 0.f;
    if (d < DQ)
      for (int h = 0; h < HQ; ++h) acc = fmaf(cg[h], V1_W[(size_t)h * DQ + d], acc);
    ws[OFF_G2 + (size_t)b * 1040 + d] = acc;
  }
  if (tid == 0) {
    float acc = 0.f;
    for (int h = 0; h < HQ; ++h) acc = fmaf(cg[h], V1_b[h], acc);
    ws[OFF_C0 + b] = acc;
  }
}

// ---------------------------------------------------------------------------
// Kernel 1: WMMA f32 GEMM.
//   base[r, 0:64]   = wenc_n[r,:] @ ws_W1[:,2067:]^T + ws_b1
//   base[r, 64:128] = wenc_n[r,:] @ we_W1[:,2067:]^T + we_b1
//   attn_logits[r]  = g2[b,:] . wenc_n[r,:] + c0[b]    (fused into same K-loop)
// One wave per 16-row tile; 8 accumulator tiles cover all 128 columns.
// A layout (16x4 f32): lanes 0-15 -> K=0/1 in v0/v1; lanes 16-31 -> K=2/3.
// C layout: VGPR i, lanes 0-15 -> (M=i, N=lane); lanes 16-31 -> (M=i+8, N=lane-16).
// ---------------------------------------------------------------------------
__global__ __launch_bounds__(256) void base_gemm_kernel(
    const float* __restrict__ wenc_n,
    const float* __restrict__ ws_W1, const float* __restrict__ ws_b1,
    const float* __restrict__ we_W1, const float* __restrict__ we_b1,
    float* __restrict__ ws)
{
  const int lane  = threadIdx.x & 31;
  const int wv    = threadIdx.x >> 5;
  const int tile  = blockIdx.x * 8 + wv;      // 0..1023
  const int row0  = tile * 16;
  const int mrow  = lane & 15;
  const int khalf = lane >> 4;                // 0: K pair {0,1}, 1: {2,3}
  const int r     = row0 + mrow;
  const int b     = r >> 10;
  const int n     = r & (NN - 1);
  const float* arow = wenc_n + ((size_t)b * NN + n) * DQ;
  const float* g2   = ws + OFF_G2 + (size_t)b * 1040;

  const float* wp[8];
  v8f acc[8];
#pragma unroll
  for (int j = 0; j < 8; ++j) {
    const int  wcol = (j * 16 + mrow) & 63;
    const float* W   = (j < 4) ? ws_W1 : we_W1;
    const float* bia = (j < 4) ? ws_b1 : we_b1;
    wp[j] = W + (size_t)wcol * DIN + EMBW;
    const float bv = bia[wcol];   // bias depends only on column -> same in all 8 C regs
    v8f c;
#pragma unroll
    for (int i = 0; i < 8; ++i) c[i] = bv;
    acc[j] = c;
  }

  float plog = 0.f;
  for (int k = 0; k < 1036; k += 4) {          // full steps: covers K 0..1035
    const int k0 = k + 2 * khalf;
    v2f A; A[0] = arow[k0]; A[1] = arow[k0 + 1];
    plog = fmaf(A[0], g2[k0], fmaf(A[1], g2[k0 + 1], plog));
#pragma unroll
    for (int j = 0; j < 8; ++j) {
      v2f Bv; Bv[0] = wp[j][k0]; Bv[1] = wp[j][k0 + 1];
      acc[j] = __builtin_amdgcn_wmma_f32_16x16x4_f32(
          false, A, false, Bv, (short)0, acc[j], false, false);
    }
  }
  { // tail: K = 1036,1037,1038 (zero-pad K=1039)
    const int k0 = 1036 + 2 * khalf;
    const bool ok1 = (k0 + 1) < DQ;
    v2f A; A[0] = arow[k0]; A[1] = ok1 ? arow[k0 + 1] : 0.f;
    plog = fmaf(A[0], g2[k0], plog);
    if (ok1) plog = fmaf(A[1], g2[k0 + 1], plog);
#pragma unroll
    for (int j = 0; j < 8; ++j) {
      v2f Bv; Bv[0] = wp[j][k0]; Bv[1] = ok1 ? wp[j][k0 + 1] : 0.f;
      acc[j] = __builtin_amdgcn_wmma_f32_16x16x4_f32(
          false, A, false, Bv, (short)0, acc[j], false, false);
    }
  }

  // combine the two lane-half logit partials (K 0/1 half + K 2/3 half)
  const float tot = plog + __shfl_xor(plog, 16, 32);
  if (khalf == 0)
    ws[OFF_LOG + b * NN + n] = tot + ws[OFF_C0 + b];

#pragma unroll
  for (int j = 0; j < 8; ++j) {
    const int col = j * 16 + mrow;
#pragma unroll
    for (int i = 0; i < 8; ++i) {
      const int rl = i + 8 * khalf;
      ws[OFF_BASE + (size_t)(row0 + rl) * 128 + col] = acc[j][i];
    }
  }
}

// ---------------------------------------------------------------------------
// block reductions (blockDim == 1024)
// ---------------------------------------------------------------------------
__device__ __forceinline__ int block_argmax(float v, int tid, float* sv, int* si) {
  sv[tid] = v; si[tid] = tid; __syncthreads();
  for (int off = 512; off > 0; off >>= 1) {
    if (tid < off) {
      const float v2 = sv[tid + off]; const int i2 = si[tid + off];
      if (v2 > sv[tid] || (v2 == sv[tid] && i2 < si[tid])) { sv[tid] = v2; si[tid] = i2; }
    }
    __syncthreads();
  }
  const int r = si[0]; __syncthreads();
  return r;
}
__device__ __forceinline__ float block_max(float v, int tid, float* sv) {
  sv[tid] = v; __syncthreads();
  for (int off = 512; off > 0; off >>= 1) {
    if (tid < off) sv[tid] = fmaxf(sv[tid], sv[tid + off]);
    __syncthreads();
  }
  const float r = sv[0]; __syncthreads();
  return r;
}
__device__ __forceinline__ float block_sum(float v, int tid, float* sv) {
  sv[tid] = v; __syncthreads();
  for (int off = 512; off > 0; off >>= 1) {
    if (tid < off) sv[tid] += sv[tid + off];
    __syncthreads();
  }
  const float r = sv[0]; __syncthreads();
  return r;
}

// ---------------------------------------------------------------------------
// Kernel 2 (per idx): span-softmax -> emb_c -> gated emb -> rank-1 update
//   add[b, h(s)] = emb . ws_W1[h, 0:2067], add[b, 64+h] = emb . we_W1[h, 0:2067]
// One block (1024 threads) per batch.
// ---------------------------------------------------------------------------
__global__ __launch_bounds__(1024) void emb_kernel(
    const float* __restrict__ wenc_n, const float* __restrict__ wenc_hs,
    const float* __restrict__ ws_W1, const float* __restrict__ we_W1,
    const int* __restrict__ wn, const int* __restrict__ wc,
    const float* __restrict__ dout, float* __restrict__ ws, int idx)
{
  __shared__ float sval[1024];
  __shared__ int   sidx[1024];
  __shared__ float a_sh[1024];
  __shared__ float emb_sh[EMBW + 1];
  const int b = blockIdx.x, tid = threadIdx.x;
  const bool gate = wn[b] > idx;

  float e0 = 0.f, e1 = 0.f;      // emb_c at d = tid and d = tid + 1024
  if (idx > 0) {
    const float* prev = dout + (size_t)(b * MLW + (idx - 1)) * NN * 2;
    const int start = block_argmax(prev[2 * tid + 0], tid, sval, sidx);
    const float me  = (tid >= start) ? prev[2 * tid + 1] : NEGV;
    const int endn  = block_argmax(me, tid, sval, sidx);
    const bool span = (tid >= start) && (tid <= endn);
    const float lg  = span ? ws[OFF_LOG + b * NN + tid] : -INFINITY;
    const float mx  = block_max(lg, tid, sval);
    const float ev  = span ? __expf(lg - mx) : 0.f;
    const float Z   = block_sum(ev, tid, sval);
    a_sh[tid] = ev / Z;
    __syncthreads();
    // emb_c[d] = sum_n a[n] * wenc_n[b,n,d]  (coalesced across threads per n)
    const float* base = wenc_n + (size_t)b * NN * DQ;
    {
      float acc = 0.f;
      const float* col = base + tid;
      for (int nn = 0; nn < NN; ++nn) acc = fmaf(a_sh[nn], col[(size_t)nn * DQ], acc);
      e0 = acc;
    }
    if (tid + 1024 < DQ) {
      float acc = 0.f;
      const float* col = base + tid + 1024;
      for (int nn = 0; nn < NN; ++nn) acc = fmaf(a_sh[nn], col[(size_t)nn * DQ], acc);
      e1 = acc;
    }
  }

  // emb = gate ? [h_sel (1028) | emb_c (1039)] : 0
  const int head = wc[b * MLW + idx];
  for (int j = tid; j < DH; j += 1024)
    emb_sh[j] = gate ? wenc_hs[((size_t)b * NHEADS + head) * DH + j] : 0.f;
  if (tid < DQ)        emb_sh[DH + tid]        = gate ? e0 : 0.f;
  if (tid + 1024 < DQ) emb_sh[DH + tid + 1024] = gate ? e1 : 0.f;
  __syncthreads();

  if (tid < 128) {
    const int h = tid & 63;
    const float* wr = ((tid < 64) ? ws_W1 : we_W1) + (size_t)h * DIN;
    float acc = 0.f;
    for (int d = 0; d < EMBW; ++d) acc = fmaf(emb_sh[d], wr[d], acc);
    ws[OFF_ADD + b * 128 + tid] = acc;
  }
}

// ---------------------------------------------------------------------------
// Kernel 3 (per idx): s/e = tanh(base + add) @ W2 + b2, masked by l_n
// ---------------------------------------------------------------------------
__global__ __launch_bounds__(256) void final_kernel(
    const float* __restrict__ ws,
    const float* __restrict__ ws_W2, const float* __restrict__ ws_b2,
    const float* __restrict__ we_W2, const float* __restrict__ we_b2,
    const int* __restrict__ l_n, float* __restrict__ dout, int idx)
{
  const int r = blockIdx.x * blockDim.x + threadIdx.x;   // 0..16383
  const int b = r >> 10, n = r & (NN - 1);
  const float* base = ws + OFF_BASE + (size_t)r * 128;
  const float* add  = ws + OFF_ADD + b * 128;
  float s = ws_b2[0], e = we_b2[0];
  for (int h = 0; h < HS; ++h) {
    s = fmaf(tanhf(base[h]      + add[h]),      ws_W2[h], s);
    e = fmaf(tanhf(base[64 + h] + add[64 + h]), we_W2[h], e);
  }
  const bool valid = n < l_n[b];
  const size_t o = ((size_t)(b * MLW + idx) * NN + n) * 2;
  dout[o]     = valid ? s : NEGV;
  dout[o + 1] = valid ? e : NEGV;
}

// ---------------------------------------------------------------------------
extern "C" void kernel_launch(void* const* d_in, const int* in_sizes, int n_in,
                              void* d_out, int out_size, void* d_ws, size_t ws_size,
                              hipStream_t stream) {
  const float* wenc_n  = (const float*)d_in[0];
  const float* wenc_hs = (const float*)d_in[1];
  const float* U1_W    = (const float*)d_in[2];
  const float* U1_b    = (const float*)d_in[3];
  const float* V1_W    = (const float*)d_in[4];
  const float* V1_b    = (const float*)d_in[5];
  const float* ws_W1   = (const float*)d_in[6];
  const float* ws_b1   = (const float*)d_in[7];
  const float* ws_W2   = (const float*)d_in[8];
  const float* ws_b2   = (const float*)d_in[9];
  const float* we_W1   = (const float*)d_in[10];
  const float* we_b1   = (const float*)d_in[11];
  const float* we_W2   = (const float*)d_in[12];
  const float* we_b2   = (const float*)d_in[13];
  const int*   l_n     = (const int*)d_in[14];
  // d_in[15] (l_hs) unused by the reference
  const int*   wn      = (const int*)d_in[16];
  const int*   wc      = (const int*)d_in[17];
  float* ws  = (float*)d_ws;
  float* out = (float*)d_out;

  init_kernel<<<NB, 256, 0, stream>>>(wenc_n, U1_W, U1_b, V1_W, V1_b, ws);
  base_gemm_kernel<<<128, 256, 0, stream>>>(wenc_n, ws_W1, ws_b1, we_W1, we_b1, ws);
  for (int idx = 0; idx < MLW; ++idx) {
    emb_kernel<<<NB, 1024, 0, stream>>>(wenc_n, wenc_hs, ws_W1, we_W1,
                                        wn, wc, out, ws, idx);
    final_kernel<<<64, 256, 0, stream>>>(ws, ws_W2, ws_b2, we_W2, we_b2,
                                         l_n, out, idx);
  }
}